// PDFA_6536940224940
// MI455X (gfx1250) — compile-verified
//
#include <hip/hip_runtime.h>
#include <hip/hip_bf16.h>

// PDFA forward: N=128 sequences, L=2048 steps, C=8 symbols, S=16 states.
// Chain product of 16x16 f32 matrices done with V_WMMA_F32_16X16X4_F32
// (4 x K=4 chunks per 16^3 product). All matrix operands staged in LDS so
// A-layout / B-layout gathers are plain ds loads (no cross-lane shuffles).

typedef __attribute__((ext_vector_type(2))) float v2f;
typedef __attribute__((ext_vector_type(8))) float v8f;

#define NSEQ 128
#define LSTEPS 2048
#define SEGS 64          // segments per sequence
#define SEG_STEPS 32     // steps per segment (one wave per segment)

// ---- WMMA helpers -------------------------------------------------------
// A-layout (16x4 f32): lane m (0-15): v0=A[m][k0+0], v1=A[m][k0+1];
//                      lane m+16   : v0=A[m][k0+2], v1=A[m][k0+3]
// B-layout (4x16 f32): lane n (0-15): v0=B[k0+0][n], v1=B[k0+1][n];
//                      lane n+16   : v0=B[k0+2][n], v1=B[k0+3][n]
// D-layout (16x16 f32): vgpr r: lanes0-15 -> D[r][n], lanes16-31 -> D[r+8][n]

__device__ inline v8f mat16_mul_lds(const float* __restrict__ A,
                                    const float* __restrict__ B) {
  const unsigned lane = threadIdx.x & 31u;
  const unsigned hf   = lane >> 4;
  const unsigned m    = lane & 15u;
  v8f acc = {};
#pragma unroll
  for (int k = 0; k < 4; ++k) {
    const int c0 = 4 * k + 2 * (int)hf;
    v2f a, b;
    a.x = A[m * 16 + c0];
    a.y = A[m * 16 + c0 + 1];
    b.x = B[(c0 + 0) * 16 + m];
    b.y = B[(c0 + 1) * 16 + m];
    acc = __builtin_amdgcn_wmma_f32_16x16x4_f32(false, a, false, b,
                                                (short)0, acc, false, false);
  }
  return acc;
}

__device__ inline void store_d_lds(float* __restrict__ D, v8f acc) {
  const unsigned lane = threadIdx.x & 31u;
  const unsigned hf   = lane >> 4;
  const unsigned m    = lane & 15u;
#pragma unroll
  for (int r = 0; r < 8; ++r)
    D[(r + 8 * hf) * 16 + m] = acc[r];
}

// ---- Kernel A: parameter preprocessing ---------------------------------
// ws layout (floats): [0,2048)  Tmat[c][p*16+n] = softmax_n(logit_trans[p,c,:])
//                     [2048,2064) init softmax
//                     [2064,2080) accepting log-sigmoid
//                     [2080,...)  segment products: NSEQ * SEGS * 256
__global__ __launch_bounds__(128) void pdfa_prep(
    const float* __restrict__ logit_init, const float* __restrict__ logit_trans,
    const float* __restrict__ logit_acc, float* __restrict__ ws) {
  const int t = threadIdx.x;
  if (t < 128) {
    const int p = t >> 3, c = t & 7;
    const float* row = logit_trans + (p * 8 + c) * 16;
    float mx = row[0];
    for (int n = 1; n < 16; ++n) mx = fmaxf(mx, row[n]);
    float e[16], sum = 0.f;
    for (int n = 0; n < 16; ++n) { e[n] = __expf(row[n] - mx); sum += e[n]; }
    const float inv = 1.f / sum;
    for (int n = 0; n < 16; ++n) ws[c * 256 + p * 16 + n] = e[n] * inv;
  }
  if (t < 16) {
    float mx = logit_init[0];
    for (int i = 1; i < 16; ++i) mx = fmaxf(mx, logit_init[i]);
    float sum = 0.f;
    for (int i = 0; i < 16; ++i) sum += __expf(logit_init[i] - mx);
    ws[2048 + t] = __expf(logit_init[t] - mx) / sum;
    const float x = logit_acc[t];
    ws[2064 + t] = fminf(x, 0.f) - log1pf(__expf(-fabsf(x)));  // log_sigmoid
  }
}

// ---- Kernel B: per-segment chain product (one wave per 32 steps) -------
__global__ __launch_bounds__(32) void pdfa_segment(
    const float* __restrict__ logx, const float* __restrict__ ws,
    float* __restrict__ segprod) {
  __shared__ float Tm[2048];               // Tmat[c][entry]
  __shared__ float xbuf[SEG_STEPS * 8];    // exp(log_input_probs) for segment
  __shared__ float mats[SEG_STEPS][256];   // row-major 16x16 per step

  const int b    = blockIdx.x / SEGS;
  const int seg  = blockIdx.x % SEGS;
  const int lane = threadIdx.x;
  const unsigned hf = lane >> 4;
  const unsigned m  = lane & 15u;

  for (int i = lane; i < 2048; i += 32) Tm[i] = ws[i];
  const float* src = logx + ((size_t)b * LSTEPS + (size_t)seg * SEG_STEPS) * 8;
  for (int i = lane; i < SEG_STEPS * 8; i += 32) xbuf[i] = __expf(src[i]);
  __syncthreads();

  // Phase 1: per_step[l][p*16+n] = sum_c x[l][c] * Tmat[c][p*16+n]
  // 16 steps per group as WMMA M-dim; K=8 via two 16x16x4 chunks; 16 N-tiles.
  for (int g = 0; g < 2; ++g) {
    v2f a[2];
#pragma unroll
    for (int k = 0; k < 2; ++k) {
      const int c0 = 4 * k + 2 * (int)hf;
      a[k].x = xbuf[(g * 16 + m) * 8 + c0];
      a[k].y = xbuf[(g * 16 + m) * 8 + c0 + 1];
    }
    for (int t = 0; t < 16; ++t) {
      v8f acc = {};
#pragma unroll
      for (int k = 0; k < 2; ++k) {
        const int c0 = 4 * k + 2 * (int)hf;
        v2f bb;
        bb.x = Tm[(c0 + 0) * 256 + t * 16 + m];
        bb.y = Tm[(c0 + 1) * 256 + t * 16 + m];
        acc = __builtin_amdgcn_wmma_f32_16x16x4_f32(false, a[k], false, bb,
                                                    (short)0, acc, false, false);
      }
#pragma unroll
      for (int r = 0; r < 8; ++r)
        mats[g * 16 + r + 8 * hf][t * 16 + m] = acc[r];
    }
  }
  __syncthreads();

  // Phase 2: in-order pairwise tree product 32 -> 1 (order-preserving).
  for (int n = SEG_STEPS / 2; n >= 1; n >>= 1) {
    for (int i = 0; i < n; ++i) {
      v8f p = mat16_mul_lds(&mats[2 * i][0], &mats[2 * i + 1][0]);
      store_d_lds(&mats[i][0], p);
    }
    __syncthreads();
  }

  float* dst = segprod + ((size_t)b * SEGS + seg) * 256;
  for (int i = lane; i < 256; i += 32) dst[i] = mats[0][i];
}

// ---- Kernel C1: combine 8 adjacent segment products -> slot j*8 --------
__global__ __launch_bounds__(32) void pdfa_combine8(float* __restrict__ segprod) {
  __shared__ float mats[8][256];
  const int b = blockIdx.x >> 3;
  const int j = blockIdx.x & 7;
  const int lane = threadIdx.x;
  float* base = segprod + ((size_t)b * SEGS + (size_t)j * 8) * 256;
  for (int i = lane; i < 8 * 256; i += 32) mats[i >> 8][i & 255] = base[i];
  __syncthreads();
  for (int n = 4; n >= 1; n >>= 1) {
    for (int i = 0; i < n; ++i) {
      v8f p = mat16_mul_lds(&mats[2 * i][0], &mats[2 * i + 1][0]);
      store_d_lds(&mats[i][0], p);
    }
    __syncthreads();
  }
  for (int i = lane; i < 256; i += 32) base[i] = mats[0][i];
}

// ---- Kernel C2: final combine + acceptance log-prob --------------------
__global__ __launch_bounds__(32) void pdfa_final(
    const float* __restrict__ segprod, const float* __restrict__ ws,
    float* __restrict__ out) {
  __shared__ float mats[8][256];
  __shared__ float red[16];
  const int b = blockIdx.x;
  const int lane = threadIdx.x;
  for (int k = 0; k < 8; ++k) {
    const float* src = segprod + ((size_t)b * SEGS + (size_t)k * 8) * 256;
    for (int i = lane; i < 256; i += 32) mats[k][i] = src[i];
  }
  __syncthreads();
  for (int n = 4; n >= 1; n >>= 1) {
    for (int i = 0; i < n; ++i) {
      v8f p = mat16_mul_lds(&mats[2 * i][0], &mats[2 * i + 1][0]);
      store_d_lds(&mats[i][0], p);
    }
    __syncthreads();
  }
  if (lane < 16) {
    float s = 0.f;
    for (int p = 0; p < 16; ++p) s += ws[2048 + p] * mats[0][p * 16 + lane];
    red[lane] = logf(s + 1e-20f) + ws[2064 + lane];
  }
  __syncthreads();
  if (lane == 0) {
    float mx = red[0];
    for (int i = 1; i < 16; ++i) mx = fmaxf(mx, red[i]);
    float sum = 0.f;
    for (int i = 0; i < 16; ++i) sum += __expf(red[i] - mx);
    out[b] = mx + logf(sum);
  }
}

extern "C" void kernel_launch(void* const* d_in, const int* in_sizes, int n_in,
                              void* d_out, int out_size, void* d_ws, size_t ws_size,
                              hipStream_t stream) {
  const float* logx = (const float*)d_in[0];   // (N,L,C) f32
  const float* li   = (const float*)d_in[1];   // (S,)
  const float* lt   = (const float*)d_in[2];   // (S,C,S)
  const float* la   = (const float*)d_in[3];   // (S,)
  float* wsf  = (float*)d_ws;
  float* outp = (float*)d_out;                 // (1,N) f32
  float* segprod = wsf + 2080;                 // NSEQ*SEGS*256 floats (~8MB)

  pdfa_prep<<<1, 128, 0, stream>>>(li, lt, la, wsf);
  pdfa_segment<<<NSEQ * SEGS, 32, 0, stream>>>(logx, wsf, segprod);
  pdfa_combine8<<<NSEQ * 8, 32, 0, stream>>>(segprod);
  pdfa_final<<<NSEQ, 32, 0, stream>>>(segprod, wsf, outp);
}